// CPSpatialAttention_103079215458
// MI455X (gfx1250) — compile-verified
//
#include <hip/hip_runtime.h>
#include <hip/hip_bf16.h>

typedef float v2f __attribute__((ext_vector_type(2)));
typedef float v8f __attribute__((ext_vector_type(8)));

#define NPATCH 8192        // B*M = 8*1024
#define P 49
#define C 64
#define PP 2401            // 49*49
#define OUT_N 25690112     // 8*64*224*224
#define SC_N 19668992      // 8192*2401

// ---- CDNA5 async global->LDS gather (guarded; falls back to plain loads) ----
#if defined(__has_builtin)
# if __has_builtin(__builtin_amdgcn_global_load_async_to_lds_b32) && \
     __has_builtin(__builtin_amdgcn_s_wait_asynccnt)
#  define USE_ASYNC_LDS 1
# endif
#endif
#ifndef USE_ASYNC_LDS
# define USE_ASYNC_LDS 0
#endif

#if USE_ASYNC_LDS
typedef __attribute__((address_space(1))) int* as1_i32p;   // global
typedef __attribute__((address_space(3))) int* as3_i32p;   // LDS

__device__ __forceinline__ void async_copy_f32(const float* g, float* l) {
    __builtin_amdgcn_global_load_async_to_lds_b32(
        (as1_i32p)g, (as3_i32p)l, 0, 0);
}
#endif

// ---------------------------------------------------------------------------
// Stage 1: per-patch G = FM*FM^T via fp32 WMMA; emit Sc (softmax) and cov_big
// ---------------------------------------------------------------------------
__global__ __launch_bounds__(128) void stage1_kernel(const float* __restrict__ x,
                                                     float* __restrict__ sc,
                                                     float* __restrict__ covb) {
    __shared__ float fm[64][68];   // FM padded (rows 49..63 zero)
    __shared__ float G[64][65];
    __shared__ float mu[64];

    const int tid = threadIdx.x;
    const int n = blockIdx.x;
    const int b = n >> 10, m = n & 1023, hb = m >> 5, wb = m & 31;
    const float* xb = x + (size_t)b * C * 224 * 224;

    // load FM[p][c] = x[b, c, hb*7+ph, wb*7+pw]; (c,p) ordering for coalescing
    for (int e = tid; e < P * C; e += 128) {
        int c = e / P, p = e - c * P;
        int ph = p / 7, pw = p - ph * 7;
        const float* gp = &xb[((size_t)c * 224 + (hb * 7 + ph)) * 224 + (wb * 7 + pw)];
#if USE_ASYNC_LDS
        async_copy_f32(gp, &fm[p][c]);
#else
        fm[p][c] = *gp;
#endif
    }
    for (int e = tid; e < 15 * 64; e += 128)
        fm[P + (e >> 6)][e & 63] = 0.0f;
#if USE_ASYNC_LDS
    __builtin_amdgcn_s_wait_asynccnt(0);
#endif
    __syncthreads();

    if (tid < P) {
        float s = 0.0f;
        for (int c = 0; c < C; ++c) s += fm[tid][c];
        mu[tid] = s * (1.0f / 64.0f);
    }

    // ---- WMMA: G(64x64 padded) = FM(64x64) * FM^T, K = 64 ----
    const int lane = tid & 31, w = tid >> 5;
    const int rlo  = lane & 15;
    const int khi  = (lane >> 4) << 1;        // 0 or 2 (K sub-offset)
    const int rowA = w * 16 + rlo;            // wave w owns tile-row w
    const int mbase = w * 16 + ((lane >> 4) << 3);

    for (int j = 0; j < 4; ++j) {
        const int colB = j * 16 + rlo;
        v8f acc = {};
        for (int k0 = 0; k0 < 64; k0 += 4) {
            v2f a, bf;
            a.x  = fm[rowA][k0 + khi];  a.y  = fm[rowA][k0 + khi + 1];
            bf.x = fm[colB][k0 + khi];  bf.y = fm[colB][k0 + khi + 1]; // B[k][n]=FM[n][k]
            acc = __builtin_amdgcn_wmma_f32_16x16x4_f32(
                false, a, false, bf, (short)0, acc, false, false);
        }
        for (int r = 0; r < 8; ++r)
            G[mbase + r][j * 16 + rlo] = acc[r];
    }
    __syncthreads();

    // cov_big[p][q] = G[p][q]/C - mu_p*mu_q   (from raw G, before softmax)
    const size_t base = (size_t)n * PP;
    for (int e = tid; e < PP; e += 128) {
        int p = e / P, q = e - p * P;
        covb[base + e] = G[p][q] * (1.0f / 64.0f) - mu[p] * mu[q];
    }
    __syncthreads();

    // softmax rows in place (rows/cols >= 49 are exact zeros, never read)
    if (tid < P) {
        float mx = -3.402823e38f;
        for (int q = 0; q < P; ++q) mx = fmaxf(mx, G[tid][q]);
        float s = 0.0f;
        for (int q = 0; q < P; ++q) { float ev = __expf(G[tid][q] - mx); G[tid][q] = ev; s += ev; }
        float inv = 1.0f / s;
        for (int q = 0; q < P; ++q) G[tid][q] *= inv;
    }
    __syncthreads();

    for (int e = tid; e < PP; e += 128) {
        int p = e / P, q = e - p * P;
        sc[base + e] = G[p][q];
    }
}

// ---------------------------------------------------------------------------
// cov_global = mean_n cov_big  (two-level: 8 n-chunks atomically accumulate)
// ---------------------------------------------------------------------------
__global__ void zero_ws_kernel(float* wscov) {
    for (int i = threadIdx.x; i < PP; i += 256) wscov[i] = 0.0f;
}

__global__ __launch_bounds__(128) void reduce_cov_kernel(const float* __restrict__ covb,
                                                         float* __restrict__ wscov) {
    const int g = (blockIdx.x % 19) * 128 + threadIdx.x;   // pq index
    const int chunk = blockIdx.x / 19;                     // 0..7
    if (g < PP) {
        float s = 0.0f;
        size_t base = (size_t)chunk * 1024 * PP + g;
        for (int i = 0; i < 1024; ++i) s += covb[base + (size_t)i * PP];
        atomicAdd(&wscov[g], s * (1.0f / 8192.0f));
    }
}

// ---------------------------------------------------------------------------
// CP-ALS rank-8, 5 iterations on 1x49x49 tensor -> cov_cp (49x49)
// ---------------------------------------------------------------------------
__device__ __forceinline__ float u01_hash(unsigned s) {
    s ^= s >> 17; s *= 0xed5ad4bbu; s ^= s >> 11; s *= 0xac4c1b51u;
    s ^= s >> 15; s *= 0x31848babu; s ^= s >> 14;
    return (float)(s >> 8) * (1.0f / 16777216.0f);
}

__device__ void invert8(const float* V, float* VI) {
    float M[8][16];
    for (int i = 0; i < 8; ++i)
        for (int j = 0; j < 8; ++j) { M[i][j] = V[i * 8 + j]; M[i][8 + j] = (i == j) ? 1.0f : 0.0f; }
    for (int c = 0; c < 8; ++c) {
        int piv = c; float mx = fabsf(M[c][c]);
        for (int r = c + 1; r < 8; ++r) { float a = fabsf(M[r][c]); if (a > mx) { mx = a; piv = r; } }
        if (piv != c)
            for (int j = 0; j < 16; ++j) { float t = M[c][j]; M[c][j] = M[piv][j]; M[piv][j] = t; }
        float d = M[c][c];
        float inv = (fabsf(d) > 1e-30f) ? (1.0f / d) : 0.0f;
        for (int j = 0; j < 16; ++j) M[c][j] *= inv;
        for (int r = 0; r < 8; ++r) if (r != c) {
            float f = M[r][c];
            for (int j = 0; j < 16; ++j) M[r][j] -= f * M[c][j];
        }
    }
    for (int i = 0; i < 8; ++i)
        for (int j = 0; j < 8; ++j) VI[i * 8 + j] = M[i][8 + j];
}

__global__ __launch_bounds__(64) void cp_als_kernel(const float* __restrict__ wscov,
                                                    float* __restrict__ wscp) {
    __shared__ float cov[PP];
    __shared__ float f0[8], f1[392], f2[392];
    __shared__ float A1[64], A2[64], V[64], VI[64];
    __shared__ float g[392], m0[8];
    const int t = threadIdx.x;

    for (int e = t; e < PP; e += 64) cov[e] = wscov[e];
    if (t < 8) f0[t] = u01_hash(1000u + t);
    for (int e = t; e < 392; e += 64) { f1[e] = u01_hash(2000u + e); f2[e] = u01_hash(3000u + e); }
    __syncthreads();

    for (int it = 0; it < 5; ++it) {
        // ---- mode 0 ----
        { int r = t >> 3, s = t & 7; float a = 0.0f, b2 = 0.0f;
          for (int p = 0; p < P; ++p) { a += f1[p*8+r]*f1[p*8+s]; b2 += f2[p*8+r]*f2[p*8+s]; }
          A1[t] = a; A2[t] = b2; V[t] = a * b2; }
        for (int e = t; e < 392; e += 64) { int q = e >> 3, r = e & 7; float s = 0.0f;
            for (int p = 0; p < P; ++p) s += cov[p*P+q] * f1[p*8+r]; g[e] = s; }
        __syncthreads();
        if (t < 8) { float s = 0.0f; for (int q = 0; q < P; ++q) s += g[q*8+t] * f2[q*8+t]; m0[t] = s; }
        __syncthreads();
        if (t == 0) invert8(V, VI);
        __syncthreads();
        if (t < 8) { float s = 0.0f; for (int r = 0; r < 8; ++r) s += m0[r] * VI[r*8+t]; f0[t] = s; }
        __syncthreads();

        // ---- mode 1 ----  V = outer(f0) .* (f2^T f2)
        { int r = t >> 3, s = t & 7; V[t] = f0[r] * f0[s] * A2[t]; }
        for (int e = t; e < 392; e += 64) { int p = e >> 3, r = e & 7; float s = 0.0f;
            for (int q = 0; q < P; ++q) s += cov[p*P+q] * f2[q*8+r]; g[e] = s * f0[r]; }
        __syncthreads();
        if (t == 0) invert8(V, VI);
        __syncthreads();
        for (int e = t; e < 392; e += 64) { int p = e >> 3, rp = e & 7; float s = 0.0f;
            for (int r = 0; r < 8; ++r) s += g[p*8+r] * VI[r*8+rp]; f1[e] = s; }
        __syncthreads();

        // ---- mode 2 ----  V = outer(f0) .* (f1^T f1) with NEW f1
        { int r = t >> 3, s = t & 7; float a = 0.0f;
          for (int p = 0; p < P; ++p) a += f1[p*8+r]*f1[p*8+s];
          A1[t] = a; V[t] = f0[r] * f0[s] * a; }
        for (int e = t; e < 392; e += 64) { int q = e >> 3, r = e & 7; float s = 0.0f;
            for (int p = 0; p < P; ++p) s += cov[p*P+q] * f1[p*8+r]; g[e] = s * f0[r]; }
        __syncthreads();
        if (t == 0) invert8(V, VI);
        __syncthreads();
        for (int e = t; e < 392; e += 64) { int q = e >> 3, rp = e & 7; float s = 0.0f;
            for (int r = 0; r < 8; ++r) s += g[q*8+r] * VI[r*8+rp]; f2[e] = s; }
        __syncthreads();
    }

    // cov_cp[p][q] = sum_r f1[p,r] * f0[r] * f2[q,r]
    for (int e = t; e < PP; e += 64) {
        int p = e / P, q = e - p * P; float s = 0.0f;
        for (int r = 0; r < 8; ++r) s += f1[p*8+r] * f0[r] * f2[q*8+r];
        wscp[e] = s;
    }
}

// ---------------------------------------------------------------------------
// Stage 4: Ec = (Sc + cov_cp) * FM via fp32 WMMA; fold; out = x*(beta*Ec + x)
// ---------------------------------------------------------------------------
__global__ __launch_bounds__(128) void stage4_kernel(const float* __restrict__ x,
                                                     const float* __restrict__ beta,
                                                     const float* __restrict__ sc,
                                                     const float* __restrict__ covcp,
                                                     float* __restrict__ outp,
                                                     float* __restrict__ ecmap) {
    __shared__ float fm[64][68];
    __shared__ float L[64][68];
    __shared__ float E[64][68];

    const int tid = threadIdx.x;
    const int n = blockIdx.x;
    const int b = n >> 10, m = n & 1023, hb = m >> 5, wb = m & 31;
    const float* xb = x + (size_t)b * C * 224 * 224;
    const size_t base = (size_t)n * PP;

    for (int e = tid; e < 4096; e += 128) L[e >> 6][e & 63] = 0.0f;
    for (int e = tid; e < P * C; e += 128) {
        int c = e / P, p = e - c * P;
        int ph = p / 7, pw = p - ph * 7;
        const float* gp = &xb[((size_t)c * 224 + (hb * 7 + ph)) * 224 + (wb * 7 + pw)];
#if USE_ASYNC_LDS
        async_copy_f32(gp, &fm[p][c]);
#else
        fm[p][c] = *gp;
#endif
    }
    for (int e = tid; e < 15 * 64; e += 128)
        fm[P + (e >> 6)][e & 63] = 0.0f;
#if USE_ASYNC_LDS
    __builtin_amdgcn_s_wait_asynccnt(0);
#endif
    __syncthreads();
    for (int e = tid; e < PP; e += 128) {
        int p = e / P, q = e - p * P;
        L[p][q] = sc[base + e] + covcp[e];
    }
    __syncthreads();

    // ---- WMMA: E(64x64) = L(64x64) * FM(64x64), K = q-dim ----
    const int lane = tid & 31, w = tid >> 5;
    const int rlo  = lane & 15;
    const int khi  = (lane >> 4) << 1;
    const int rowA = w * 16 + rlo;
    const int mbase = w * 16 + ((lane >> 4) << 3);

    for (int j = 0; j < 4; ++j) {
        const int colB = j * 16 + rlo;
        v8f acc = {};
        for (int k0 = 0; k0 < 64; k0 += 4) {
            v2f a, bf;
            a.x  = L[rowA][k0 + khi];      a.y  = L[rowA][k0 + khi + 1];
            bf.x = fm[k0 + khi][colB];     bf.y = fm[k0 + khi + 1][colB]; // B[k][c]=FM[k][c]
            acc = __builtin_amdgcn_wmma_f32_16x16x4_f32(
                false, a, false, bf, (short)0, acc, false, false);
        }
        for (int r = 0; r < 8; ++r)
            E[mbase + r][j * 16 + rlo] = acc[r];
    }
    __syncthreads();

    // fold + elementwise; out/ecmap are write-only streams -> non-temporal
    // stores so x (re-read) and Sc stay resident in the 192MB L2.
    const float bv = beta[0];
    for (int e = tid; e < P * C; e += 128) {
        int c = e / P, p = e - c * P;
        int ph = p / 7, pw = p - ph * 7;
        size_t addr = ((size_t)(b * C + c) * 224 + (hb * 7 + ph)) * 224 + (wb * 7 + pw);
        float ec = E[p][c];
        float xv = fm[p][c];
        __builtin_nontemporal_store(ec, &ecmap[addr]);
        __builtin_nontemporal_store(xv * (bv * ec + xv), &outp[addr]);
    }
}

// ---------------------------------------------------------------------------
extern "C" void kernel_launch(void* const* d_in, const int* in_sizes, int n_in,
                              void* d_out, int out_size, void* d_ws, size_t ws_size,
                              hipStream_t stream) {
    const float* x    = (const float*)d_in[0];
    const float* beta = (const float*)d_in[1];

    float* outp  = (float*)d_out;                 // [8,64,224,224]
    float* sc    = outp + OUT_N;                  // [8192,49,49]
    float* covb  = sc + SC_N;                     // [8192,49,49]
    float* ecmap = covb + SC_N;                   // [8,64,224,224]

    float* ws_cov = (float*)d_ws;                 // 2401 (+pad)
    float* ws_cp  = ws_cov + 2432;                // 2401

    stage1_kernel<<<NPATCH, 128, 0, stream>>>(x, sc, covb);
    zero_ws_kernel<<<1, 256, 0, stream>>>(ws_cov);
    reduce_cov_kernel<<<19 * 8, 128, 0, stream>>>(covb, ws_cov);
    cp_als_kernel<<<1, 64, 0, stream>>>(ws_cov, ws_cp);
    stage4_kernel<<<NPATCH, 128, 0, stream>>>(x, beta, sc, ws_cp, outp, ecmap);
}